// DTWCVAELoss_5145370820977
// MI455X (gfx1250) — compile-verified
//
#include <hip/hip_runtime.h>
#include <math.h>

#define B_SZ  128
#define T_SZ  256
#define L_SZ  128
#define BIGV  100000000.0f
#define SGAMMA 0.1f
#define INVG   10.0f
#define BW     150
#define RPT   8          // rows (ii) per lane; 32 lanes * 8 = 256 rows

typedef __attribute__((ext_vector_type(2))) float v2f;
typedef __attribute__((ext_vector_type(8))) float v8f;

// ---------------------------------------------------------------------------
// Banded soft-DTW, one (pair, batch) per wave32. Each lane owns 8 consecutive
// DP rows; prev/prev2 rows live in registers. The only cross-lane dependency
// per anti-diagonal is the neighbor lane's last cell -> 2 x __shfl_up.
// No barriers in the hot loop; y tile staged once in LDS.
// ---------------------------------------------------------------------------
__global__ __launch_bounds__(32) void sdtw_kernel(
    const float* __restrict__ pred, const float* __restrict__ gt,
    const int* __restrict__ perm_gen, const int* __restrict__ perm_real,
    float* __restrict__ partial /* [4*B] */)
{
  __shared__ float sy[T_SZ * 3];

  const int b    = blockIdx.x;
  const int pair = blockIdx.y;
  const int tid  = threadIdx.x;   // 0..31

  const float* x;
  const float* y;
  if (pair == 0)      { x = pred + b * T_SZ * 3; y = gt   + b * T_SZ * 3; }
  else if (pair == 1) { x = gt   + b * T_SZ * 3; y = pred + b * T_SZ * 3; }
  else if (pair == 2) { x = pred + b * T_SZ * 3; y = pred + perm_gen[b]  * T_SZ * 3; }
  else                { x = gt   + b * T_SZ * 3; y = gt   + perm_real[b] * T_SZ * 3; }

  // gfx1250 global_prefetch on the y trajectory before the LDS stage
  __builtin_prefetch(y + tid * 24, 0, 0);

  // stage y (256x3 f32) into LDS: 24 floats per lane, lane stride 32
  for (int i = tid; i < T_SZ * 3; i += 32) sy[i] = y[i];
  __syncthreads();   // single-wave WG: LDS ops are in-order within the wave

  // this lane's 8 x-rows in registers (ci = tid*8 + k)
  float x0[RPT], x1[RPT], x2[RPT];
#pragma unroll
  for (int k = 0; k < RPT; ++k) {
    const int ci = tid * RPT + k;
    x0[k] = x[ci * 3 + 0];
    x1[k] = x[ci * 3 + 1];
    x2[k] = x[ci * 3 + 2];
  }

  float pv[RPT], pv2[RPT], nr[RPT];
#pragma unroll
  for (int k = 0; k < RPT; ++k) { pv[k] = BIGV; pv2[k] = BIGV; }

  const int ii0 = tid * RPT + 1;   // first row index owned by this lane

  for (int d = 2; d <= 2 * T_SZ; ++d) {
    // neighbor lane's boundary cells (prev[ii0-1], prev2[ii0-1])
    float upN = __shfl_up(pv[RPT - 1], 1);
    float dgN = __shfl_up(pv2[RPT - 1], 1);
    if (tid == 0) {
      upN = BIGV;                          // prev[0] is always BIG
      dgN = (d == 2) ? 0.0f : BIGV;        // prev2[0] = 0 only at DP origin
    }
#pragma unroll
    for (int k = 0; k < RPT; ++k) {
      const int ii = ii0 + k;
      const int jj = d - ii;
      const float up = (k == 0) ? upN : pv[k - 1];
      const float lf = pv[k];
      const float dg = (k == 0) ? dgN : pv2[k - 1];
      float r = BIGV;
      int dij = ii - jj; if (dij < 0) dij = -dij;
      if ((jj >= 1) && (jj <= T_SZ) && (dij <= BW)) {
        // stable soft-min, gamma = 0.1 (matches jax logsumexp)
        const float m = fminf(up, fminf(lf, dg));
        const float s = __expf((m - up) * INVG) + __expf((m - lf) * INVG) +
                        __expf((m - dg) * INVG);
        const float smin = m - SGAMMA * __logf(s);
        const int cj = jj - 1;
        const float dd = fabsf(x0[k] - sy[cj * 3 + 0]) +
                         fabsf(x1[k] - sy[cj * 3 + 1]) +
                         fabsf(x2[k] - sy[cj * 3 + 2]);
        const float e = 1.0f - __expf(-dd);
        r = 0.6f * e * e * dd + smin;      // focal-weighted Manhattan + softmin
      }
      nr[k] = r;
    }
#pragma unroll
    for (int k = 0; k < RPT; ++k) { pv2[k] = pv[k]; pv[k] = nr[k]; }
  }

  // answer = last[:, T] at d = 2T -> row ii = 256 = lane 31, sub-cell 7
  if (tid == 31) partial[pair * B_SZ + b] = pv[RPT - 1];
}

// ---------------------------------------------------------------------------
// Finalize: KL via V_WMMA_F32_16X16X4_F32 (ones-matvec reduction over L=128),
// transition terms, and all means -> 5 scalar outputs.
// ---------------------------------------------------------------------------
__device__ __forceinline__ float block_sum(float v, float* red) {
  const int tid = threadIdx.x;
  __syncthreads();
  red[tid] = v;
  __syncthreads();
  for (int s = 128; s > 0; s >>= 1) {
    if (tid < s) red[tid] += red[tid + s];
    __syncthreads();
  }
  const float r = red[0];
  __syncthreads();
  return r;
}

__global__ __launch_bounds__(256) void finalize_kernel(
    const float* __restrict__ partial,
    const float* __restrict__ mu, const float* __restrict__ lv,
    const float* __restrict__ pred, const float* __restrict__ gt,
    const float* __restrict__ ptc, float* __restrict__ out)
{
  __shared__ float red[256];
  __shared__ float klbuf[B_SZ];

  const int tid  = threadIdx.x;
  const int lane = tid & 31;
  const int wave = tid >> 5;
  const int m0   = wave * 16;                 // each wave reduces 16 batches
  const int m    = m0 + (lane & 15);          // A-matrix row = lane (16x4 f32 layout)
  const int koff = (lane >> 4) * 2;           // VGPR0: K=0|2, VGPR1: K=1|3

  const float* murow = mu + m * L_SZ;
  const float* lvrow = lv + m * L_SZ;

  v2f bones; bones.x = 1.0f; bones.y = 1.0f;  // B = ones -> row-sum reduction
  v8f c = {};
  for (int kk = 0; kk < L_SZ / 4; ++kk) {
    const int l0 = kk * 4 + koff;
    const float lv0 = lvrow[l0],     mu0 = murow[l0];
    const float lv1 = lvrow[l0 + 1], mu1 = murow[l0 + 1];
    v2f a;
    a.x = 1.0f + lv0 - mu0 * mu0 - __expf(lv0);
    a.y = 1.0f + lv1 - mu1 * mu1 - __expf(lv1);
    c = __builtin_amdgcn_wmma_f32_16x16x4_f32(
        /*neg_a=*/false, a, /*neg_b=*/false, bones,
        /*c_mod=*/(short)0, c, /*reuse_a=*/false, /*reuse_b=*/false);
  }
  // D[m][n] == sum_l t[m][l] for every n; lanes 0/16 hold N=0 for M=v / v+8
  if (lane == 0) {
#pragma unroll
    for (int v = 0; v < 8; ++v) klbuf[m0 + v] = c[v];
  } else if (lane == 16) {
#pragma unroll
    for (int v = 0; v < 8; ++v) klbuf[m0 + 8 + v] = c[v];
  }
  __syncthreads();

  float s0 = 0.f, s1 = 0.f, s2 = 0.f, s3 = 0.f, klb = 0.f, auxb = 0.f, trb = 0.f;
  if (tid < B_SZ) {
    const int b = tid;
    s0 = partial[b];
    s1 = partial[B_SZ + b];
    s2 = partial[2 * B_SZ + b];
    s3 = partial[3 * B_SZ + b];

    const float kld = -0.5f * klbuf[b];
    klb = fmaxf(kld - 0.5f, 0.0f);            // free-bits KL

    const float* gp = gt + b * T_SZ * 3;
    const float* pp = pred + b * T_SZ * 3;
    float prevg = gp[2];
    float prevs = 1.0f / (1.0f + __expf(-(pp[2] - 0.5f) * 10.0f));
    float gtr = 0.0f, pts = 0.0f;
    for (int t = 1; t < T_SZ; ++t) {
      const float g = gp[t * 3 + 2];
      gtr += fabsf(g - prevg); prevg = g;
      const float sg = 1.0f / (1.0f + __expf(-(pp[t * 3 + 2] - 0.5f) * 10.0f));
      pts += fabsf(sg - prevs); prevs = sg;
    }
    const float da = ptc[b] - gtr; auxb = da * da;
    const float dt = pts - gtr;    trb = dt * dt;
  }

  const float sum0 = block_sum(s0, red);
  const float sum1 = block_sum(s1, red);
  const float sum2 = block_sum(s2, red);
  const float sum3 = block_sum(s3, red);
  const float sumk = block_sum(klb, red);
  const float suma = block_sum(auxb, red);
  const float sumt = block_sum(trb, red);

  if (tid == 0) {
    const float inv = 1.0f / (float)B_SZ;
    const float sim = sum0 * inv + sum1 * inv;
    const float nag = sim + fabsf(sum2 * inv - sum3 * inv);
    const float kl  = sumk * inv;
    const float aux = suma * inv;
    const float trg = sumt * inv;
    const float total = nag + 1.0f * kl + 0.1f * aux + 0.5f * trg;
    out[0] = total; out[1] = nag; out[2] = kl; out[3] = aux; out[4] = trg;
  }
}

// ---------------------------------------------------------------------------
extern "C" void kernel_launch(void* const* d_in, const int* in_sizes, int n_in,
                              void* d_out, int out_size, void* d_ws, size_t ws_size,
                              hipStream_t stream) {
  (void)in_sizes; (void)n_in; (void)out_size; (void)ws_size;
  const float* pred = (const float*)d_in[0];
  const float* gt   = (const float*)d_in[1];
  const float* mu   = (const float*)d_in[2];
  const float* lv   = (const float*)d_in[3];
  const float* ptc  = (const float*)d_in[4];
  const int*   pg   = (const int*)d_in[5];
  const int*   pr   = (const int*)d_in[6];
  float* out = (float*)d_out;
  float* partial = (float*)d_ws;   // 4*B floats

  sdtw_kernel<<<dim3(B_SZ, 4), 32, 0, stream>>>(pred, gt, pg, pr, partial);
  finalize_kernel<<<1, 256, 0, stream>>>(partial, mu, lv, pred, gt, ptc, out);
}